// xLSTM_2869038153890
// MI455X (gfx1250) — compile-verified
//
#include <hip/hip_runtime.h>
#include <cmath>

typedef float v2f __attribute__((ext_vector_type(2)));
typedef float v8f __attribute__((ext_vector_type(8)));

constexpr int CB = 128, CS = 256, CF = 10, CE = 40, CNH = 8, CDH = 5;
constexpr int CI = 128, CK = 4, CNQK = 32;
constexpr int ROWS = CB * CS; // 32768

__device__ __forceinline__ float sigmoidf_(float x){ return 1.f/(1.f+expf(-x)); }
__device__ __forceinline__ float siluf_(float x){ return x/(1.f+expf(-x)); }
__device__ __forceinline__ float logsigf_(float x){
  return (x >= 0.f) ? -log1pf(expf(-x)) : (x - log1pf(expf(x)));
}

// ---------------- input projection + positional encoding ----------------
__global__ void k_inproj(const float* __restrict__ x, const float* __restrict__ w,
                         const float* __restrict__ b, float* __restrict__ h){
  int idx = blockIdx.x*blockDim.x + threadIdx.x;
  if (idx >= ROWS*CE) return;
  int row = idx / CE, e = idx % CE;
  int s = row % CS;
  float acc = b[e];
  #pragma unroll
  for (int f=0; f<CF; ++f) acc += x[row*CF+f]*w[e*CF+f];
  int ebase = e & ~1;
  float div = expf(-logf(10000.f) * (float)ebase / (float)CE);
  float ang = (float)s * div;
  acc += (e & 1) ? cosf(ang) : sinf(ang);
  h[idx] = acc;
}

// ---------------- layernorm (per row, D elements) ----------------
__global__ void k_layernorm(const float* __restrict__ x, const float* __restrict__ w,
                            float* __restrict__ y, int D){
  int row = blockIdx.x*blockDim.x + threadIdx.x;
  if (row >= ROWS) return;
  const float* xr = x + (size_t)row*D;
  float mu = 0.f;
  for (int i=0;i<D;++i) mu += xr[i];
  mu /= (float)D;
  float var = 0.f;
  for (int i=0;i<D;++i){ float d = xr[i]-mu; var += d*d; }
  var /= (float)D;
  float rs = rsqrtf(var + 1e-5f);
  float* yr = y + (size_t)row*D;
  for (int i=0;i<D;++i) yr[i] = (xr[i]-mu)*rs*w[i];
}

// ---------------- generic f32 WMMA GEMM: C[M,N] = A[M,Kd] * W[N,Kd]^T ----------------
// per-wave 16x16 output tile; chained v_wmma_f32_16x16x4_f32 over Kd/4 chunks.
// OOB on N handled branch-free (clamped row + select) so no EXEC churn around WMMAs.
__global__ __launch_bounds__(256) void k_gemm(const float* __restrict__ A, int lda,
                                              const float* __restrict__ W,
                                              float* __restrict__ C, int ldc,
                                              int M, int N, int Kd, int acc){
  int lane = threadIdx.x & 31, wid = threadIdx.x >> 5;
  int tileM = blockIdx.x*8 + wid;
  if (tileM*16 >= M) return;
  int tileN = blockIdx.y;
  int m16 = lane & 15;
  int khalf = (lane >> 4) << 1;      // 0 or 2
  int wr = tileN*16 + m16;           // weight row (output column)
  float wmask = (wr < N) ? 1.f : 0.f;
  int wrc = (wr < N) ? wr : (N-1);   // clamped: load always in-bounds
  const float* arow = A + (size_t)(tileM*16 + m16)*lda;
  const float* wrow = W + (size_t)wrc*Kd;
  v8f c;
  #pragma unroll
  for (int r=0;r<8;++r) c[r] = 0.f;
  int nk = Kd >> 2;
  for (int kk=0; kk<nk; ++kk){
    int colk = kk*4 + khalf;
    v2f a, bf;
    a.x  = arow[colk];           a.y  = arow[colk+1];
    bf.x = wrow[colk]*wmask;     bf.y = wrow[colk+1]*wmask;
    c = __builtin_amdgcn_wmma_f32_16x16x4_f32(false, a, false, bf, (short)0, c, false, false);
  }
  int col = tileN*16 + m16;
  if (col < N){
    #pragma unroll
    for (int r=0;r<8;++r){
      int row = tileM*16 + r + ((lane>>4)<<3);
      size_t o = (size_t)row*ldc + col;
      if (acc) C[o] += c[r]; else C[o] = c[r];
    }
  }
}

// ---------------- causal depthwise conv (K=4) + SiLU ----------------
__global__ void k_convsilu(const float* __restrict__ xin, const float* __restrict__ w,
                           const float* __restrict__ bias, float* __restrict__ y,
                           int Cc, int ldx, int ldy){
  int idx = blockIdx.x*blockDim.x + threadIdx.x;
  if (idx >= ROWS*Cc) return;
  int row = idx / Cc, c = idx % Cc;
  int b = row / CS, s = row % CS;
  float acc = bias[c];
  #pragma unroll
  for (int k=0;k<CK;++k){
    int sp = s - 3 + k;
    if (sp >= 0) acc += xin[(size_t)(b*CS+sp)*ldx + c]*w[c*CK+k];
  }
  y[(size_t)row*ldy + c] = siluf_(acc);
}

// ---------------- block-diagonal 4x4 headwise q,k,v ----------------
__global__ void k_headwise(const float* __restrict__ xc, const float* __restrict__ up,
                           const float* __restrict__ qw, const float* __restrict__ kw,
                           const float* __restrict__ vw, float* __restrict__ qkv){
  int idx = blockIdx.x*blockDim.x + threadIdx.x;
  if (idx >= ROWS*3*CI) return;
  int row = idx / (3*CI), j = idx % (3*CI);
  int sel = j / CI, c = j % CI;
  int blk = c >> 2, o = c & 3;
  const float* wsel = (sel==0)? qw : (sel==1)? kw : vw;
  float acc = 0.f;
  #pragma unroll
  for (int i=0;i<4;++i){
    float xv = (sel==2) ? up[(size_t)row*(2*CI) + blk*4 + i]      // v from x_m
                        : xc[(size_t)row*CI + blk*4 + i];         // q,k from x_c
    acc += xv * wsel[blk*16 + o*4 + i];
  }
  qkv[(size_t)row*(3*CI) + j] = acc;
}

// ---------------- gate GEMVs: ip, fp in (B, NH, S) layout ----------------
__global__ void k_gates(const float* __restrict__ qkv,
                        const float* __restrict__ igw, const float* __restrict__ igb,
                        const float* __restrict__ fgw, const float* __restrict__ fgb,
                        float* __restrict__ ip, float* __restrict__ fp){
  int idx = blockIdx.x*blockDim.x + threadIdx.x;
  if (idx >= ROWS*CNH) return;
  int row = idx / CNH, hh = idx % CNH;
  int b = row / CS, s = row % CS;
  const float* g = qkv + (size_t)row*(3*CI);
  const float* wi = igw + hh*(3*CI);
  const float* wf = fgw + hh*(3*CI);
  float ai = igb[hh], af = fgb[hh];
  for (int j=0;j<3*CI;++j){ float gv = g[j]; ai += gv*wi[j]; af += gv*wf[j]; }
  size_t o = (size_t)(b*CNH + hh)*CS + s;
  ip[o] = ai; fp[o] = af;
}

// ---------------- per (b,h) scans: Fc = cumsum(logsig(fp)); mD = Fc + runmax(ip - Fc) ----------------
__global__ __launch_bounds__(256) void k_scan(const float* __restrict__ ip, const float* __restrict__ fp,
                                              float* __restrict__ Fc, float* __restrict__ mD){
  __shared__ float sd[CS];
  int bh = blockIdx.x, s = threadIdx.x;
  float ls = logsigf_(fp[(size_t)bh*CS + s]);
  sd[s] = ls; __syncthreads();
  for (int off=1; off<CS; off<<=1){
    float t = (s >= off) ? sd[s-off] : 0.f;
    __syncthreads();
    sd[s] += t;
    __syncthreads();
  }
  float fc = sd[s];
  Fc[(size_t)bh*CS + s] = fc;
  float v = ip[(size_t)bh*CS + s] - fc;
  __syncthreads();
  sd[s] = v; __syncthreads();
  for (int off=1; off<CS; off<<=1){
    float t = (s >= off) ? sd[s-off] : -INFINITY;
    __syncthreads();
    sd[s] = fmaxf(sd[s], t);
    __syncthreads();
  }
  mD[(size_t)bh*CS + s] = fc + sd[s];
}

// ---------------- single-pass decay attention with WMMA ----------------
__global__ __launch_bounds__(256) void k_attention(const float* __restrict__ qkv,
                                                   const float* __restrict__ ip,
                                                   const float* __restrict__ Fc,
                                                   const float* __restrict__ mD,
                                                   float* __restrict__ att){
  __shared__ float Qs[CS*16], Ks[CS*16], Vs[CS*16];
  __shared__ float Fcs[CS], ips[CS], mDs[CS];
  __shared__ float stage[8*16*17];
  int bh = blockIdx.x; int b = bh / CNH; int h = bh % CNH;
  int tid = threadIdx.x;
  for (int i = tid; i < CS*16; i += 256){
    int s = i >> 4, d = i & 15;
    const float* rowp = qkv + (size_t)(b*CS + s)*(3*CI) + h*16 + d;
    Qs[i] = rowp[0]; Ks[i] = rowp[CI]; Vs[i] = rowp[2*CI];
  }
  Fcs[tid] = Fc[(size_t)bh*CS + tid];
  ips[tid] = ip[(size_t)bh*CS + tid];
  mDs[tid] = mD[(size_t)bh*CS + tid];
  __syncthreads();

  int lane = tid & 31, wid = tid >> 5;
  int m16 = lane & 15;
  int khalf = (lane >> 4) << 1;
  int mrow8 = (lane >> 4) << 3;
  float* st = stage + wid*16*17;

  for (int si = 0; si < 2; ++si){
    int stile = wid + si*8;
    v8f Hacc; float rs[8];
    #pragma unroll
    for (int r=0;r<8;++r){ Hacc[r] = 0.f; rs[r] = 0.f; }

    for (int tt = 0; tt <= stile; ++tt){
      // ---- qk tile: Q(16x16) x K^T(16x16) via 4 chained 16x16x4 WMMAs ----
      v8f c;
      #pragma unroll
      for (int r=0;r<8;++r) c[r] = 0.f;
      #pragma unroll
      for (int kk=0; kk<4; ++kk){
        int col = kk*4 + khalf;
        v2f a, bf;
        a.x  = Qs[(stile*16 + m16)*16 + col];  a.y  = Qs[(stile*16 + m16)*16 + col + 1];
        bf.x = Ks[(tt*16 + m16)*16 + col];     bf.y = Ks[(tt*16 + m16)*16 + col + 1];
        c = __builtin_amdgcn_wmma_f32_16x16x4_f32(false, a, false, bf, (short)0, c, false, false);
      }
      // ---- apply decay matrix D elementwise, accumulate row sums, stage C tile ----
      int t_glob = tt*16 + m16;        // column index within seq
      float fct = Fcs[t_glob], ipt = ips[t_glob];
      #pragma unroll
      for (int r=0;r<8;++r){
        int m = r + mrow8;
        int s_glob = stile*16 + m;
        float val = 0.f;
        if (t_glob <= s_glob)
          val = c[r]*0.25f*expf(Fcs[s_glob] - fct + ipt - mDs[s_glob]);
        rs[r] += val;
        st[m*17 + m16] = val;          // wave-private staging
      }
      asm volatile("s_wait_dscnt 0x0" ::: "memory");  // LDS stores visible to wave's loads
      // ---- H tile accumulate: C(16x16) x V(16x16) via 4 chained WMMAs ----
      #pragma unroll
      for (int kk=0; kk<4; ++kk){
        int col = kk*4 + khalf;
        v2f a, bf;
        a.x  = st[m16*17 + col];           a.y  = st[m16*17 + col + 1];
        bf.x = Vs[(tt*16 + col)*16 + m16]; bf.y = Vs[(tt*16 + col + 1)*16 + m16];
        Hacc = __builtin_amdgcn_wmma_f32_16x16x4_f32(false, a, false, bf, (short)0, Hacc, false, false);
      }
    }
    // ---- reduce row sums across the 16-lane half, normalize and store ----
    #pragma unroll
    for (int mask=1; mask<16; mask<<=1){
      #pragma unroll
      for (int r=0;r<8;++r) rs[r] += __shfl_xor(rs[r], mask, 32);
    }
    #pragma unroll
    for (int r=0;r<8;++r){
      int m = r + mrow8;
      int s_glob = stile*16 + m;
      float nrm = fmaxf(fabsf(rs[r]), expf(-mDs[s_glob]));
      att[(size_t)(b*CS + s_glob)*CI + h*16 + m16] = Hacc[r] / (nrm + 1e-6f);
    }
  }
}

// ---------------- multihead norm (16/head) + skip*x_c, then * silu(z) ----------------
__global__ void k_mhn_fuse(const float* __restrict__ att, const float* __restrict__ mhnw,
                           const float* __restrict__ skip, const float* __restrict__ xc,
                           const float* __restrict__ up, float* __restrict__ din){
  int idx = blockIdx.x*blockDim.x + threadIdx.x;
  if (idx >= ROWS*CNH) return;
  int row = idx / CNH, hh = idx % CNH;
  const float* a = att + (size_t)row*CI + hh*16;
  float mu = 0.f;
  for (int i=0;i<16;++i) mu += a[i];
  mu *= (1.f/16.f);
  float var = 0.f;
  for (int i=0;i<16;++i){ float d = a[i]-mu; var += d*d; }
  var *= (1.f/16.f);
  float rsq = rsqrtf(var + 1e-5f);
  for (int i=0;i<16;++i){
    int c = hh*16 + i;
    float mh = (a[i]-mu)*rsq*mhnw[c];
    float zv = up[(size_t)row*(2*CI) + CI + c];
    din[(size_t)row*CI + c] = (mh + skip[c]*xc[(size_t)row*CI + c]) * siluf_(zv);
  }
}

// ---------------- sLSTM gates (block-diagonal 5x5 per head) ----------------
__global__ void k_slstm_gates(const float* __restrict__ xc2, const float* __restrict__ xn,
                              const float* __restrict__ wi, const float* __restrict__ wf,
                              const float* __restrict__ wz, const float* __restrict__ wo,
                              float* __restrict__ gx){
  int idx = blockIdx.x*blockDim.x + threadIdx.x;
  if (idx >= ROWS*4*CE) return;
  int row = idx / (4*CE), rem = idx % (4*CE);
  int g = rem / CE, e = rem % CE;
  int hh = e / CDH, d = e % CDH;
  const float* src = (g < 2) ? xc2 : xn;
  const float* w = (g==0)? wi : (g==1)? wf : (g==2)? wz : wo;
  float acc = 0.f;
  #pragma unroll
  for (int i=0;i<CDH;++i)
    acc += src[(size_t)row*CE + hh*CDH + i] * w[hh*25 + d*5 + i];
  gx[(size_t)g*ROWS*CE + (size_t)row*CE + e] = acc;
}

// ---------------- sLSTM sequential scan: one wave per (b,h) ----------------
__global__ __launch_bounds__(32) void k_slstm_scan(const float* __restrict__ gx,
                                                   const float* __restrict__ R,
                                                   const float* __restrict__ bias,
                                                   float* __restrict__ hs){
  int bh = blockIdx.x; int b = bh / CNH; int hh = bh % CNH;
  int lane = threadIdx.x;
  int g = lane / CDH, d = lane % CDH;
  bool active = lane < 20;
  float Rreg[CDH]; float breg = 0.f;
  #pragma unroll
  for (int e=0;e<CDH;++e) Rreg[e] = 0.f;
  if (active){
    #pragma unroll
    for (int e=0;e<CDH;++e) Rreg[e] = R[((hh*4 + g)*CDH + d)*CDH + e];
    breg = bias[(hh*4 + g)*CDH + d];
  }
  float hst=0.f, cst=0.f, nst=0.f, mst=0.f;
  const size_t RT = (size_t)ROWS*CE;
  for (int s=0;s<CS;++s){
    float hp0 = __shfl(hst, 0, 32), hp1 = __shfl(hst, 1, 32), hp2 = __shfl(hst, 2, 32);
    float hp3 = __shfl(hst, 3, 32), hp4 = __shfl(hst, 4, 32);
    float raw = 0.f;
    if (active){
      raw = gx[(size_t)g*RT + (size_t)(b*CS+s)*CE + hh*CDH + d] + breg
          + Rreg[0]*hp0 + Rreg[1]*hp1 + Rreg[2]*hp2 + Rreg[3]*hp3 + Rreg[4]*hp4;
    }
    float i_r = __shfl(raw, d,      32);
    float f_r = __shfl(raw, 5  + d, 32);
    float z_r = __shfl(raw, 10 + d, 32);
    float o_r = __shfl(raw, 15 + d, 32);
    if (lane < CDH){
      float lfm = mst + logsigf_(f_r);
      float mnew = fmaxf(i_r, lfm);
      float ig = expf(i_r - mnew);
      float fg = expf(lfm - mnew);
      cst = fg*cst + ig*tanhf(z_r);
      nst = fg*nst + ig;
      mst = mnew;
      hst = sigmoidf_(o_r)*cst/nst;
      hs[(size_t)(b*CS+s)*CE + hh*CDH + lane] = hst;
    }
  }
}

// ---------------- multihead norm (5/head) added into h ----------------
__global__ void k_mhn_add(const float* __restrict__ hsv, const float* __restrict__ w,
                          float* __restrict__ h){
  int idx = blockIdx.x*blockDim.x + threadIdx.x;
  if (idx >= ROWS*CNH) return;
  int row = idx / CNH, hh = idx % CNH;
  const float* a = hsv + (size_t)row*CE + hh*CDH;
  float mu = 0.f;
  for (int i=0;i<CDH;++i) mu += a[i];
  mu *= (1.f/CDH);
  float var = 0.f;
  for (int i=0;i<CDH;++i){ float dd = a[i]-mu; var += dd*dd; }
  var *= (1.f/CDH);
  float rsq = rsqrtf(var + 1e-5f);
  for (int i=0;i<CDH;++i)
    h[(size_t)row*CE + hh*CDH + i] += (a[i]-mu)*rsq*w[hh*CDH + i];
}

// ---------------- GEGLU: gelu(gpre)*up ----------------
__global__ void k_geglu(const float* __restrict__ ff1, float* __restrict__ ffd){
  int idx = blockIdx.x*blockDim.x + threadIdx.x;
  if (idx >= ROWS*64) return;
  int row = idx / 64, j = idx % 64;
  float gp = ff1[(size_t)row*128 + j];
  float upv = ff1[(size_t)row*128 + 64 + j];
  float gelu = 0.5f*gp*(1.f + erff(gp*0.70710678118654752f));
  ffd[(size_t)row*64 + j] = gelu*upv;
}

// ---------------- final layernorm + scalar head ----------------
__global__ void k_out(const float* __restrict__ h, const float* __restrict__ lnw,
                      const float* __restrict__ ow, const float* __restrict__ ob,
                      float* __restrict__ out){
  int row = blockIdx.x*blockDim.x + threadIdx.x;
  if (row >= ROWS) return;
  const float* xr = h + (size_t)row*CE;
  float mu = 0.f;
  for (int i=0;i<CE;++i) mu += xr[i];
  mu /= (float)CE;
  float var = 0.f;
  for (int i=0;i<CE;++i){ float d = xr[i]-mu; var += d*d; }
  var /= (float)CE;
  float rsq = rsqrtf(var + 1e-5f);
  float acc = ob[0];
  for (int i=0;i<CE;++i) acc += (xr[i]-mu)*rsq*lnw[i]*ow[i];
  out[row] = acc;
}

extern "C" void kernel_launch(void* const* d_in, const int* in_sizes, int n_in,
                              void* d_out, int out_size, void* d_ws, size_t ws_size,
                              hipStream_t stream){
  const float* x        = (const float*)d_in[0];
  const float* inw      = (const float*)d_in[1];
  const float* inb      = (const float*)d_in[2];
  const float* m_ln_w   = (const float*)d_in[3];
  const float* m_up_w   = (const float*)d_in[4];
  const float* m_conv_w = (const float*)d_in[5];
  const float* m_conv_b = (const float*)d_in[6];
  const float* m_q_w    = (const float*)d_in[7];
  const float* m_k_w    = (const float*)d_in[8];
  const float* m_v_w    = (const float*)d_in[9];
  const float* m_ig_w   = (const float*)d_in[10];
  const float* m_ig_b   = (const float*)d_in[11];
  const float* m_fg_w   = (const float*)d_in[12];
  const float* m_fg_b   = (const float*)d_in[13];
  const float* m_mhn_w  = (const float*)d_in[14];
  const float* m_skip   = (const float*)d_in[15];
  const float* m_down_w = (const float*)d_in[16];
  const float* s_ln1_w  = (const float*)d_in[17];
  const float* s_conv_w = (const float*)d_in[18];
  const float* s_conv_b = (const float*)d_in[19];
  const float* s_wi     = (const float*)d_in[20];
  const float* s_wf     = (const float*)d_in[21];
  const float* s_wz     = (const float*)d_in[22];
  const float* s_wo     = (const float*)d_in[23];
  const float* s_R      = (const float*)d_in[24];
  const float* s_b      = (const float*)d_in[25];
  const float* s_mhn_w  = (const float*)d_in[26];
  const float* s_ln2_w  = (const float*)d_in[27];
  const float* s_ffu_w  = (const float*)d_in[28];
  const float* s_ffd_w  = (const float*)d_in[29];
  const float* post_ln  = (const float*)d_in[30];
  const float* out_w    = (const float*)d_in[31];
  const float* out_b    = (const float*)d_in[32];
  float* out = (float*)d_out;
  float* ws  = (float*)d_ws;

  float* hB   = ws;
  float* xnB  = hB  + (size_t)ROWS*CE;
  float* upB  = xnB + (size_t)ROWS*CE;
  float* xcB  = upB + (size_t)ROWS*(2*CI);
  float* qkvB = xcB + (size_t)ROWS*CI;
  float* ipB  = qkvB + (size_t)ROWS*(3*CI);
  float* fpB  = ipB + (size_t)CB*CNH*CS;
  float* FcB  = fpB + (size_t)CB*CNH*CS;
  float* mDB  = FcB + (size_t)CB*CNH*CS;
  float* attB = mDB + (size_t)CB*CNH*CS;
  float* dinB = attB + (size_t)ROWS*CI;
  // reuse (disjoint in time):
  float* xc2B = attB;   // ROWS*CE
  float* gxB  = qkvB;   // 4*ROWS*CE
  float* hsB  = dinB;   // ROWS*CE
  float* ff1B = upB;    // ROWS*128
  float* ffdB = xcB;    // ROWS*64

  auto cdiv = [](int a, int b){ return (a + b - 1)/b; };

  k_inproj<<<cdiv(ROWS*CE,256),256,0,stream>>>(x, inw, inb, hB);

  auto mlstm = [&](int j){
    k_layernorm<<<cdiv(ROWS,256),256,0,stream>>>(hB, m_ln_w + j*CE, xnB, CE);
    k_gemm<<<dim3(ROWS/128, (2*CI)/16),256,0,stream>>>(xnB, CE, m_up_w + (size_t)j*2*CI*CE,
                                                       upB, 2*CI, ROWS, 2*CI, CE, 0);
    k_convsilu<<<cdiv(ROWS*CI,256),256,0,stream>>>(upB, m_conv_w + j*CI*CK, m_conv_b + j*CI,
                                                   xcB, CI, 2*CI, CI);
    k_headwise<<<cdiv(ROWS*3*CI,256),256,0,stream>>>(xcB, upB, m_q_w + j*CNQK*16,
                                                     m_k_w + j*CNQK*16, m_v_w + j*CNQK*16, qkvB);
    k_gates<<<cdiv(ROWS*CNH,256),256,0,stream>>>(qkvB, m_ig_w + j*CNH*3*CI, m_ig_b + j*CNH,
                                                 m_fg_w + j*CNH*3*CI, m_fg_b + j*CNH, ipB, fpB);
    k_scan<<<CB*CNH, CS, 0, stream>>>(ipB, fpB, FcB, mDB);
    k_attention<<<CB*CNH, 256, 0, stream>>>(qkvB, ipB, FcB, mDB, attB);
    k_mhn_fuse<<<cdiv(ROWS*CNH,256),256,0,stream>>>(attB, m_mhn_w + j*CI, m_skip + j*CI,
                                                    xcB, upB, dinB);
    k_gemm<<<dim3(ROWS/128, cdiv(CE,16)),256,0,stream>>>(dinB, CI, m_down_w + (size_t)j*CE*CI,
                                                         hB, CE, ROWS, CE, CI, 1);
  };

  mlstm(0);

  // sLSTM block
  k_layernorm<<<cdiv(ROWS,256),256,0,stream>>>(hB, s_ln1_w, xnB, CE);
  k_convsilu<<<cdiv(ROWS*CE,256),256,0,stream>>>(xnB, s_conv_w, s_conv_b, xc2B, CE, CE, CE);
  k_slstm_gates<<<cdiv(ROWS*4*CE,256),256,0,stream>>>(xc2B, xnB, s_wi, s_wf, s_wz, s_wo, gxB);
  k_slstm_scan<<<CB*CNH, 32, 0, stream>>>(gxB, s_R, s_b, hsB);
  k_mhn_add<<<cdiv(ROWS*CNH,256),256,0,stream>>>(hsB, s_mhn_w, hB);

  // FF block (GEGLU)
  k_layernorm<<<cdiv(ROWS,256),256,0,stream>>>(hB, s_ln2_w, xnB, CE);
  k_gemm<<<dim3(ROWS/128, 128/16),256,0,stream>>>(xnB, CE, s_ffu_w, ff1B, 128, ROWS, 128, CE, 0);
  k_geglu<<<cdiv(ROWS*64,256),256,0,stream>>>(ff1B, ffdB);
  k_gemm<<<dim3(ROWS/128, cdiv(CE,16)),256,0,stream>>>(ffdB, 64, s_ffd_w, hB, CE, ROWS, CE, 64, 1);

  mlstm(1);
  mlstm(2);

  k_out<<<cdiv(ROWS,256),256,0,stream>>>(hB, post_ln, out_w, out_b, out);
}